// RandomConv2d_42399917146878
// MI455X (gfx1250) — compile-verified
//
#include <hip/hip_runtime.h>

// RandomConv2d via implicit-im2col GEMM on CDNA5 WMMA (f32 16x16x4).
// out[b,o,l] = sum_ck W[o,ck] * Pmask[ck,l],  ck = c*9 + (kh*3+kw).
// K-chunk = 36 = lcm(4,9): exactly 4 channels x 9 kernel positions per chunk,
// so kernel-position/mask/bounds logic is hoisted out of the K loop entirely.

typedef __attribute__((ext_vector_type(2))) float v2f;
typedef __attribute__((ext_vector_type(8))) float v8f;

#define BC    64            // input channels
#define HH    112
#define WW    112
#define OC    64            // output channels
#define CKK   576           // 64 * 9
#define LL    12544         // 112*112 output positions per batch
#define KC    36            // K-chunk (4 channels * 9 positions)
#define NCHUNK (CKK / KC)   // 16
#define LROW  44            // padded LDS row (floats): gcd(44,64)=4 -> conflict-free
                            // 16-lane b64 reads; 44*4 % 16 == 0 keeps b128 alignment

__global__ __launch_bounds__(256) void rand_conv2d_wmma_f32(
    const float* __restrict__ x,        // [B, C, H, W]
    const float* __restrict__ weight,   // [O, C, 3, 3] == [O, CKK]
    const int*   __restrict__ mask_idx, // [L]
    float*       __restrict__ out)      // [B, O, L]
{
    __shared__ __align__(16) float As[OC * LROW];  // A chunk: [m=0..63][k=0..35]
    __shared__ __align__(16) float Bs[64 * LROW];  // B chunk: [n=0..63][k=0..35]

    const int t     = threadIdx.x;
    const int nblkL = LL / 64;                 // 196 N-tiles per batch
    const int b     = blockIdx.x / nblkL;
    const int lbase = (blockIdx.x % nblkL) * 64;

    // ---- per-thread constants for B staging ----
    const int sn   = t & 63;                   // patch column this thread fills
    const int skq  = t >> 6;                   // channel phase 0..3 within chunk
    const int sl   = lbase + sn;               // global patch index
    const int soh  = sl / WW;
    const int sow  = sl - soh * WW;
    const int smid = mask_idx[sl];             // masked kernel position for this patch

    // base pointer at (c = skq, ih = soh-1, iw = sow-1); OOB only when predicated off
    const float* xptr = x + ((size_t)b * BC + skq) * (HH * WW)
                          + (soh - 1) * WW + (sow - 1);

    // 9 load predicates: bounds (PAD=1) AND not the masked kernel position.
    // All compile-time indexed below, so these live in registers/masks.
    const bool rok[3] = { soh >= 1, true, soh < HH - 1 };
    const bool cok[3] = { sow >= 1, true, sow < WW - 1 };
    bool ok[9];
    #pragma unroll
    for (int kh = 0; kh < 3; ++kh)
        #pragma unroll
        for (int kw = 0; kw < 3; ++kw)
            ok[kh * 3 + kw] = rok[kh] && cok[kw] && ((kh * 3 + kw) != smid);

    float* browp = &Bs[sn * LROW + skq * 9];   // this thread's 9 LDS slots

    // ---- per-thread constants for A staging ----
    const int am = t >> 2;                     // weight row (o)
    const int ak = t & 3;                      // interleave phase; k = ak + 4j
    const float* wrow = weight + am * CKK + ak;

    // ---- wave tile assignment ----
    const int wid   = t >> 5;                  // 8 waves
    const int lane  = t & 31;
    const int mblk  = wid & 3;                 // 16-row block of O
    const int nblk0 = (wid >> 2) << 1;         // two 16-col patch blocks
    const int nblk1 = nblk0 + 1;
    const int lrow  = lane & 15;
    const int lhi   = lane >> 4;

    v8f acc0 = {0.f,0.f,0.f,0.f,0.f,0.f,0.f,0.f};
    v8f acc1 = {0.f,0.f,0.f,0.f,0.f,0.f,0.f,0.f};

    for (int ci = 0; ci < NCHUNK; ++ci) {
        const int kc = ci * KC;
        __syncthreads();   // previous chunk fully consumed

        // prefetch next chunk's channel block toward the WGP
        if (ci + 1 < NCHUNK)
            __builtin_prefetch(xptr + 4 * (HH * WW), 0, 3);

        // ---- stage A: 64x36 weight slab (9 coalesced b32 loads/thread) ----
        #pragma unroll
        for (int j = 0; j < 9; ++j)
            As[am * LROW + ak + 4 * j] = wrow[kc + 4 * j];

        // ---- stage B: 3x3 neighborhood of channel (4*ci + skq), predicated ----
        #pragma unroll
        for (int kh = 0; kh < 3; ++kh) {
            #pragma unroll
            for (int kw = 0; kw < 3; ++kw) {
                const int m9 = kh * 3 + kw;
                float v = 0.0f;
                if (ok[m9]) v = xptr[kh * WW + kw];
                browp[m9] = v;
            }
        }
        xptr += 4 * (HH * WW);                 // next 4 channels
        __syncthreads();

        // ---- 9 WMMA k-steps over this chunk (18 WMMAs/wave) ----
        #pragma unroll
        for (int klo = 0; klo < KC; klo += 4) {
            const int koff = klo + (lhi << 1); // 2 K-values per lane
            const v2f a  = *(const v2f*)(&As[(mblk  * 16 + lrow) * LROW + koff]);
            const v2f b0 = *(const v2f*)(&Bs[(nblk0 * 16 + lrow) * LROW + koff]);
            const v2f b1 = *(const v2f*)(&Bs[(nblk1 * 16 + lrow) * LROW + koff]);
            acc0 = __builtin_amdgcn_wmma_f32_16x16x4_f32(
                false, a, false, b0, (short)0, acc0, false, false);
            acc1 = __builtin_amdgcn_wmma_f32_16x16x4_f32(
                false, a, false, b1, (short)0, acc1, false, false);
        }
    }

    // ---- epilogue: D layout (M = r + 8*(lane>>4), N = lane&15) -> coalesced ----
    const int obase = mblk * 16 + (lhi << 3);
    float* outb = out + (size_t)b * OC * LL;
    #pragma unroll
    for (int r = 0; r < 8; ++r) {
        const int o = obase + r;
        outb[(size_t)o * LL + (lbase + nblk0 * 16 + lrow)] = acc0[r];
        outb[(size_t)o * LL + (lbase + nblk1 * 16 + lrow)] = acc1[r];
    }
}

extern "C" void kernel_launch(void* const* d_in, const int* in_sizes, int n_in,
                              void* d_out, int out_size, void* d_ws, size_t ws_size,
                              hipStream_t stream) {
    const float* x      = (const float*)d_in[0];
    const float* weight = (const float*)d_in[1];
    const int*   midx   = (const int*)d_in[2];
    float*       out    = (float*)d_out;

    const int B = 8;
    dim3 grid(B * (LL / 64));   // 1568 blocks
    dim3 block(256);            // 8 wave32s
    rand_conv2d_wmma_f32<<<grid, block, 0, stream>>>(x, weight, midx, out);
}